// HATLoss_90177133346935
// MI455X (gfx1250) — compile-verified
//
#include <hip/hip_runtime.h>
#include <math.h>

typedef __attribute__((ext_vector_type(2))) float v2f;
typedef __attribute__((ext_vector_type(4))) float v4f;
typedef __attribute__((ext_vector_type(8))) float v8f;

#define PEPS 1e-5f
#define DIM  64

__device__ __forceinline__ float wave_sum32(float v) {
#pragma unroll
  for (int o = 16; o > 0; o >>= 1) v += __shfl_xor(v, o, 32);
  return v;
}

// Uniform broadcast from a compile-time-constant lane: v_readlane_b32 -> SGPR
// (avoids the ds_bpermute path that __shfl takes).
__device__ __forceinline__ float readlane_f32(float v, int srclane) {
  return __int_as_float(__builtin_amdgcn_readlane(__float_as_int(v), srclane));
}

// ---------------------------------------------------------------------------
// Kernel 0: zero rank counters + f64 accumulators (ws is not re-poisoned
// between replays, so every launch must re-initialize its own accumulators).
// ---------------------------------------------------------------------------
__global__ void hat_zero_kernel(int* __restrict__ ranks,
                                unsigned long long* __restrict__ acc, int E) {
  int i = blockIdx.x * blockDim.x + threadIdx.x;
  if (i < E) ranks[i] = 0;
  if (i < 4) acc[i] = 0ull;
}

// ---------------------------------------------------------------------------
// Kernel 1: per-node squared norms. 8 lanes per row (2x float4 each),
// shuffle-reduce within the 8-lane group. Streams the 128MB table once
// from HBM; it then stays resident in the 192MB L2 for the gather kernels.
// ---------------------------------------------------------------------------
__global__ void hat_norms_kernel(const float* __restrict__ h,
                                 float* __restrict__ norms, int N) {
  int t = blockIdx.x * blockDim.x + threadIdx.x;
  int row = t >> 3;
  int l = t & 7;
  if (row >= N) return;
  const float* p = h + (size_t)row * DIM + l * 8;
  v4f a = *(const v4f*)p;
  v4f b = *(const v4f*)(p + 4);
  float s = a.x * a.x + a.y * a.y + a.z * a.z + a.w * a.w +
            b.x * b.x + b.y * b.y + b.z * b.z + b.w * b.w;
  s += __shfl_xor(s, 4, 32);
  s += __shfl_xor(s, 2, 32);
  s += __shfl_xor(s, 1, 32);
  if (l == 0) norms[row] = s;
}

// ---------------------------------------------------------------------------
// One wave scores 16 edges with V_WMMA_F32_16X16X4_F32.
//   A (16x4)  : lane L holds u-row (L&15), VGPR0/1 = K = koff, koff+1
//   B (4x16)  : lane L holds v-col (L&15), VGPR0/1 = K = koff, koff+1
//   koff = (L<16 ? 0 : 2) + 4*step      -> identical float2 addressing
// Two accumulators (even/odd K-steps) halve the WMMA RAW chain.
// Diagonal of D (16x16 f32, 8 VGPRs) = the 16 dot products:
//   diag(m) : m<8 -> VGPR m @ lane m ; m>=8 -> VGPR m-8 @ lane m+16
// EXEC must be all ones: indices are clamped, contributions masked.
// ---------------------------------------------------------------------------
__device__ __forceinline__ float hat_score16(const float* __restrict__ h,
                                             const float* __restrict__ norms,
                                             const int* __restrict__ srcs,
                                             const int* __restrict__ dsts,
                                             long long wb, long long E,
                                             int lane, bool* valid_out) {
  int m = lane & 15;
  long long e0 = wb + m;
  bool valid = (lane < 16) && (e0 < E);
  long long e = (e0 < E) ? e0 : (E - 1);   // clamp: keep all lanes executing
  int su = srcs[e];
  int sv = dsts[e];
  int koff = (lane >> 4) << 1;             // 0 for lanes 0-15, 2 for 16-31
  const float* up = h + (size_t)su * DIM + koff;
  const float* vp = h + (size_t)sv * DIM + koff;

  v8f c0 = {0.f, 0.f, 0.f, 0.f, 0.f, 0.f, 0.f, 0.f};
  v8f c1 = {0.f, 0.f, 0.f, 0.f, 0.f, 0.f, 0.f, 0.f};
#pragma unroll
  for (int i = 0; i < 8; ++i) {
    v2f a0 = *(const v2f*)(up + 8 * i);        // K-step 2i
    v2f b0 = *(const v2f*)(vp + 8 * i);
    v2f a1 = *(const v2f*)(up + 8 * i + 4);    // K-step 2i+1
    v2f b1 = *(const v2f*)(vp + 8 * i + 4);
    c0 = __builtin_amdgcn_wmma_f32_16x16x4_f32(false, a0, false, b0,
                                               (short)0, c0, false, false);
    c1 = __builtin_amdgcn_wmma_f32_16x16x4_f32(false, a1, false, b1,
                                               (short)0, c1, false, false);
  }

  // Merge accumulators, then extract diag(D) into lanes 0..15 via
  // v_readlane broadcasts (uniform SGPRs) + cndmask selects.
  float dot = 0.f;
#pragma unroll
  for (int r = 0; r < 8; ++r) {
    float cr = c0[r] + c1[r];
    float dl = readlane_f32(cr, r);        // diag(r)
    float dh = readlane_f32(cr, r + 24);   // diag(r+8)
    dot = (lane == r) ? dl : dot;
    dot = (lane == r + 8) ? dh : dot;
  }

  float nu = norms[su];
  float nv = norms[sv];
  float sq = fmaxf(nu + nv - 2.f * dot, 0.f);       // ||u-v||^2
  float alpha = fmaxf(1.f - nu, PEPS);
  float beta  = fmaxf(1.f - nv, PEPS);
  float gamma = fmaxf(1.f + 2.f * sq / (alpha * beta), 1.f + PEPS);
  float d = acoshf(gamma);
  *valid_out = valid;
  return d * d;
}

// ---------------------------------------------------------------------------
// Kernel 2: positive edges. Store score (needed by neg pass), accumulate
// softplus(score) into f64 accumulator (score >= 0).
// ---------------------------------------------------------------------------
__global__ void hat_pos_kernel(const float* __restrict__ h,
                               const float* __restrict__ norms,
                               const int* __restrict__ src,
                               const int* __restrict__ dst,
                               float* __restrict__ pos_score,
                               double* __restrict__ acc, int E) {
  int lane = threadIdx.x & 31;
  long long wave = (long long)blockIdx.x * (blockDim.x >> 5) + (threadIdx.x >> 5);
  long long wb = wave * 16;
  bool valid;
  float score = hat_score16(h, norms, src, dst, wb, E, lane, &valid);
  if (valid) pos_score[wb + lane] = score;
  // softplus(x) for x>=0:  x + log1p(exp(-x))
  float sp = valid ? (score + log1pf(expf(-score))) : 0.f;
  sp = wave_sum32(sp);
  if (lane == 0) atomicAdd(&acc[0], (double)sp);
}

// ---------------------------------------------------------------------------
// Kernel 3: negative edges. Accumulate softplus(-score) and per-positive
// rank counts (#negs with smaller distance^2, i.e. larger logit).
// ---------------------------------------------------------------------------
__global__ void hat_neg_kernel(const float* __restrict__ h,
                               const float* __restrict__ norms,
                               const int* __restrict__ src,
                               const int* __restrict__ dst,
                               const float* __restrict__ pos_score,
                               int* __restrict__ ranks,
                               double* __restrict__ acc, long long E_neg, int nn) {
  int lane = threadIdx.x & 31;
  long long wave = (long long)blockIdx.x * (blockDim.x >> 5) + (threadIdx.x >> 5);
  long long wb = wave * 16;
  bool valid;
  float ns = hat_score16(h, norms, src, dst, wb, E_neg, lane, &valid);
  float sp = 0.f;
  if (valid) {
    long long e = wb + lane;
    int p = (int)(e / nn);
    float ps = pos_score[p];
    // softplus(-x) for x>=0: log1p(exp(-x))
    sp = log1pf(expf(-ns));
    if (ns < ps) atomicAdd(&ranks[p], 1);  // -ns > -ps  (strict)
  }
  sp = wave_sum32(sp);
  if (lane == 0) atomicAdd(&acc[1], (double)sp);
}

// ---------------------------------------------------------------------------
// Kernel 4: ranks -> sum of reciprocal ranks.
// ---------------------------------------------------------------------------
__global__ void hat_rr_kernel(const int* __restrict__ ranks,
                              double* __restrict__ acc, int E) {
  int i = blockIdx.x * blockDim.x + threadIdx.x;
  float rr = (i < E) ? 1.f / (float)(ranks[i] + 1) : 0.f;
  rr = wave_sum32(rr);
  if ((threadIdx.x & 31) == 0) atomicAdd(&acc[2], (double)rr);
}

// ---------------------------------------------------------------------------
// Kernel 5: finalize scalars.
// ---------------------------------------------------------------------------
__global__ void hat_fin_kernel(const double* __restrict__ acc,
                               float* __restrict__ out,
                               long long E_pos, long long E_neg) {
  if (blockIdx.x == 0 && threadIdx.x == 0) {
    out[0] = (float)(acc[0] / (double)E_pos + acc[1] / (double)E_neg);
    out[1] = (float)(acc[2] / (double)E_pos);
  }
}

extern "C" void kernel_launch(void* const* d_in, const int* in_sizes, int n_in,
                              void* d_out, int out_size, void* d_ws, size_t ws_size,
                              hipStream_t stream) {
  const float* h       = (const float*)d_in[0];
  const int*   pos_src = (const int*)d_in[1];
  const int*   pos_dst = (const int*)d_in[2];
  const int*   neg_src = (const int*)d_in[3];
  const int*   neg_dst = (const int*)d_in[4];

  const int       E_pos = in_sizes[1];
  const long long E_neg = in_sizes[3];
  const int       N     = in_sizes[0] / DIM;
  int nn = (E_pos > 0) ? (int)(E_neg / E_pos) : 1;
  if (nn < 1) nn = 1;

  // Workspace layout: norms[N] | pos_score[E_pos] | ranks[E_pos] | acc[4] f64
  float* norms     = (float*)d_ws;
  float* pos_score = norms + N;
  int*   ranks     = (int*)(pos_score + E_pos);
  size_t acc_off = (((size_t)N + 2ull * (size_t)E_pos) * 4ull + 15ull) & ~(size_t)15;
  double* acc = (double*)((char*)d_ws + acc_off);
  float* out = (float*)d_out;

  const int B = 256;  // 8 wave32 waves per block

  hat_zero_kernel<<<(E_pos + B - 1) / B, B, 0, stream>>>(
      ranks, (unsigned long long*)acc, E_pos);

  hat_norms_kernel<<<((size_t)N * 8 + B - 1) / B, B, 0, stream>>>(h, norms, N);

  long long waves_p = ((long long)E_pos + 15) / 16;
  hat_pos_kernel<<<(unsigned)((waves_p + 7) / 8), B, 0, stream>>>(
      h, norms, pos_src, pos_dst, pos_score, acc, E_pos);

  long long waves_n = (E_neg + 15) / 16;
  hat_neg_kernel<<<(unsigned)((waves_n + 7) / 8), B, 0, stream>>>(
      h, norms, neg_src, neg_dst, pos_score, ranks, acc, E_neg, nn);

  hat_rr_kernel<<<(E_pos + B - 1) / B, B, 0, stream>>>(ranks, acc, E_pos);

  hat_fin_kernel<<<1, 32, 0, stream>>>(acc, out, (long long)E_pos, E_neg);
}